// Mamba_90477781058076
// MI455X (gfx1250) — compile-verified
//
#include <hip/hip_runtime.h>
#include <hip/hip_bf16.h>
#include <math.h>

typedef __attribute__((ext_vector_type(2))) float        v2f;
typedef __attribute__((ext_vector_type(8))) float        v8f;
typedef __attribute__((ext_vector_type(4))) unsigned int v4u;
typedef __attribute__((ext_vector_type(8))) int          v8i;
typedef __attribute__((ext_vector_type(4))) int          v4i;

#define TILE_M 256
#define TILE_N 32
#define TILE_K 32
#define LDS_PAD 2   // must match TDM pad_amount encoding below (code 1 => 2 dwords)

__device__ __forceinline__ float silu_f(float x) {
    return x * (1.0f / (1.0f + __expf(-x)));
}
__device__ __forceinline__ float softplus_f(float x) {
    return (x > 20.0f) ? x : log1pf(__expf(x));
}

// ---------------------------------------------------------------------------
// Tensor Data Mover: DMA a 2D tile (tile_rows x 32 fp32, row stride
// row_stride_elems) from global memory into LDS at byte offset lds_addr.
// D# per ISA 8.3/8.4.  LDS padding: +2 dwords after every 32 dwords
// (pad_interval code 4 = 32 dwords, pad_amount code 1 = 2 dwords) => LDS row
// stride of TILE_K + LDS_PAD = 34 dwords, bank-conflict-free for WMMA frags.
// Tracked on TENSORcnt.
// ---------------------------------------------------------------------------
__device__ __forceinline__ void tdm_load_tile(unsigned lds_addr,
                                              const float* __restrict__ gptr,
                                              unsigned tile_rows,
                                              unsigned long long row_stride_elems)
{
    const unsigned long long ga = (unsigned long long)(uintptr_t)gptr;
    v4u g0;
    g0[0] = 1u;                                        // count=1 (valid user D#)
    g0[1] = lds_addr;                                  // LDS byte address
    g0[2] = (unsigned)(ga & 0xFFFFFFFFull);            // global_addr[31:0]
    g0[3] = (unsigned)((ga >> 32) & 0x01FFFFFFull)     // global_addr[56:32]
            | 0x80000000u;                             // type = 2 ("image")

    const unsigned tdim = 1u << 20;                    // >> any tile: no OOB clip
    v8i g1;
    g1[0] = (int)((2u << 16)        // data_size = 2 -> 4-byte elements
                | (1u << 20)        // pad_enable
                | (4u << 22)        // pad_interval code 4 -> every 32 dwords
                | (1u << 25));      // pad_amount  code 1 -> 2 dwords
    g1[1] = (int)((tdim & 0xFFFFu) << 16);                         // tensor_dim0 lo
    g1[2] = (int)((tdim >> 16) | ((tdim & 0xFFFFu) << 16));        // dim0 hi | dim1 lo
    g1[3] = (int)((tdim >> 16) | ((unsigned)TILE_K << 16));        // dim1 hi | tile_dim0=32
    g1[4] = (int)(tile_rows & 0xFFFFu);                            // tile_dim1 (tile_dim2=0)
    g1[5] = (int)(unsigned)(row_stride_elems & 0xFFFFFFFFull);     // dim0_stride lo32
    g1[6] = (int)(unsigned)((row_stride_elems >> 32) & 0xFFFFull); // dim0_stride hi16
    g1[7] = 0;

    const v4i gz = {0, 0, 0, 0};
#if defined(__clang_major__) && (__clang_major__ >= 23)
    const v8i gz8 = {0, 0, 0, 0, 0, 0, 0, 0};
    __builtin_amdgcn_tensor_load_to_lds(g0, g1, gz, gz, gz8, 0);
#else
    __builtin_amdgcn_tensor_load_to_lds(g0, g1, gz, gz, 0);
#endif
}

// ---------------------------------------------------------------------------
// C[M,N](ldc) = A[M,K](lda) @ W[N,K](ldw)^T   via V_WMMA_F32_16X16X4_F32
// EPI==1: C = softplus(C + bias[n])  (dt projection)
// Grid (N/32, M/256), 256 threads (8 waves); each wave owns a 32x32 C tile =
// four 16x16 WMMA accumulators. TDM double-buffers the A/W tiles in LDS:
// wave 0 issues tensor_load_to_lds for chunk i+1 while all waves run chunk i.
// ---------------------------------------------------------------------------
template <int EPI>
__global__ __launch_bounds__(256)
void wmma_gemm_nt(const float* __restrict__ A, int lda,
                  const float* __restrict__ W, int ldw,
                  const float* __restrict__ bias,
                  float* __restrict__ C, int ldc,
                  int K)
{
    __shared__ float As[2][TILE_M][TILE_K + LDS_PAD];
    __shared__ float Bs[2][TILE_N][TILE_K + LDS_PAD];

    const int tid  = threadIdx.x;
    const int w    = tid >> 5;     // wave 0..7 -> M sub-tile
    const int lane = tid & 31;
    const int half = lane >> 4;    // K-pair select (A) / row-pair select (B)
    const int mn   = lane & 15;    // M index in A frag, N index in B frag

    const int bm = blockIdx.y * TILE_M;
    const int bn = blockIdx.x * TILE_N;

    const unsigned ldsA[2] = { (unsigned)(uintptr_t)&As[0][0][0],
                               (unsigned)(uintptr_t)&As[1][0][0] };
    const unsigned ldsB[2] = { (unsigned)(uintptr_t)&Bs[0][0][0],
                               (unsigned)(uintptr_t)&Bs[1][0][0] };

    v8f acc00 = {}; v8f acc01 = {}; v8f acc10 = {}; v8f acc11 = {};

    // Prologue: wave 0 kicks off the DMA for chunk 0.
    if (tid < 32) {
        tdm_load_tile(ldsA[0], &A[(size_t)bm * lda], TILE_M, (unsigned long long)lda);
        tdm_load_tile(ldsB[0], &W[(size_t)bn * ldw], TILE_N, (unsigned long long)ldw);
    }

    int buf = 0;
    for (int k0 = 0; k0 < K; k0 += TILE_K, buf ^= 1) {
        if (tid < 32) {
            __builtin_amdgcn_s_wait_tensorcnt(0);   // chunk k0 landed in LDS buffer buf
        }
        __syncthreads();                            // publish LDS; WAR-safe for buf^1

        // Overlap: DMA chunk k0+32 into the other buffer during compute.
        if (tid < 32 && (k0 + TILE_K) < K) {
            tdm_load_tile(ldsA[buf ^ 1], &A[(size_t)bm * lda + k0 + TILE_K],
                          TILE_M, (unsigned long long)lda);
            tdm_load_tile(ldsB[buf ^ 1], &W[(size_t)bn * ldw + k0 + TILE_K],
                          TILE_N, (unsigned long long)ldw);
        }

        // 8 K-steps of 4 => 32 v_wmma_f32_16x16x4_f32 per wave per chunk
        #pragma unroll
        for (int kk = 0; kk < TILE_K; kk += 4) {
            v2f a0, a1, b0, b1;
            // A 16x4 frag (ISA 7.12.2): lane=half*16+m holds K=2*half,2*half+1
            a0.x = As[buf][w * 32 + mn     ][kk + 2 * half];
            a0.y = As[buf][w * 32 + mn     ][kk + 2 * half + 1];
            a1.x = As[buf][w * 32 + 16 + mn][kk + 2 * half];
            a1.y = As[buf][w * 32 + 16 + mn][kk + 2 * half + 1];
            // B 4x16 frag: lane=half*16+n holds K=2*half,2*half+1 of column n
            b0.x = Bs[buf][mn     ][kk + 2 * half];
            b0.y = Bs[buf][mn     ][kk + 2 * half + 1];
            b1.x = Bs[buf][16 + mn][kk + 2 * half];
            b1.y = Bs[buf][16 + mn][kk + 2 * half + 1];

            acc00 = __builtin_amdgcn_wmma_f32_16x16x4_f32(false, a0, false, b0, (short)0, acc00, false, false);
            acc01 = __builtin_amdgcn_wmma_f32_16x16x4_f32(false, a0, false, b1, (short)0, acc01, false, false);
            acc10 = __builtin_amdgcn_wmma_f32_16x16x4_f32(false, a1, false, b0, (short)0, acc10, false, false);
            acc11 = __builtin_amdgcn_wmma_f32_16x16x4_f32(false, a1, false, b1, (short)0, acc11, false, false);
        }
    }

    // Epilogue. C/D layout: VGPR r at lane half*16+n => row r+8*half, col n.
    v8f accs[2][2] = { { acc00, acc01 }, { acc10, acc11 } };
    #pragma unroll
    for (int mt = 0; mt < 2; ++mt) {
        #pragma unroll
        for (int nt = 0; nt < 2; ++nt) {
            const int gm0 = bm + w * 32 + mt * 16 + 8 * half;
            const int gn  = bn + nt * 16 + mn;
            float bv = 0.0f;
            if (EPI == 1) bv = bias[gn];
            #pragma unroll
            for (int r = 0; r < 8; ++r) {
                float v = accs[mt][nt][r];
                if (EPI == 1) v = softplus_f(v + bv);
                C[(size_t)(gm0 + r) * ldc + gn] = v;
            }
        }
    }
}

// ---------------------------------------------------------------------------
// Depthwise causal conv (k=4) along L + SiLU.  u_in = xz[:, :, 0:2048].
// uc[t] = sum_j u_in[t+j-3] * w[j]  (zero pad), one thread per (token, d).
// ---------------------------------------------------------------------------
__global__ __launch_bounds__(256)
void conv_silu_kernel(const float* __restrict__ xz,
                      const float* __restrict__ conv_w,
                      float* __restrict__ u)
{
    const int idx = blockIdx.x * 256 + threadIdx.x;   // [0, 4096*2048)
    const int d   = idx & 2047;
    const int tok = idx >> 11;                        // 0..4095 (= b*L)
    const int t   = tok & 2047;                       // position within L

    const float w0 = conv_w[d * 4 + 0];
    const float w1 = conv_w[d * 4 + 1];
    const float w2 = conv_w[d * 4 + 2];
    const float w3 = conv_w[d * 4 + 3];

    const float* col = xz + (size_t)tok * 4096 + d;   // u-half column
    float acc = w3 * col[0];
    if (t >= 1) acc += w2 * col[-(ptrdiff_t)4096];
    if (t >= 2) acc += w1 * col[-(ptrdiff_t)8192];
    if (t >= 3) acc += w0 * col[-(ptrdiff_t)12288];
    u[idx] = silu_f(acc);
}

// ---------------------------------------------------------------------------
// Selective scan. One lane per (channel, state): 65536 lanes = 2048 wave32s.
// 16 lanes (one channel) share dt/u loads (broadcast) and butterfly-reduce
// y_t = sum_n h_n * C_n via ds lane-permutes. Lane n==0 fuses +u*D and the
// silu(gate) multiply. yfull may alias u (read-before-write within a step).
// ---------------------------------------------------------------------------
__global__ __launch_bounds__(256)
void scan_kernel(const float* __restrict__ u,
                 const float* __restrict__ dtb,
                 const float* __restrict__ params,
                 const float* __restrict__ xz,      // gate = xz[:, :, 2048+d]
                 const float* __restrict__ log_A,
                 const float* __restrict__ Dv,
                 float* __restrict__ yfull)
{
    const int n  = threadIdx.x & 15;
    const int cg = blockIdx.x * 16 + (threadIdx.x >> 4);   // 0..4095
    const int bb = cg >> 11;
    const int d  = cg & 2047;

    const float A_dn = -__expf(log_A[d * 16 + n]);
    const float D_d  = Dv[d];

    float h = 0.0f;
    for (int t = 0; t < 2048; ++t) {
        const size_t rowud = ((size_t)(bb * 2048 + t)) * 2048 + d;
        const size_t rowp  = ((size_t)(bb * 2048 + t)) * 96;

        const float dt_t = dtb[rowud];
        const float u_t  = u[rowud];
        const float Bn   = params[rowp + 64 + n];
        const float Cn   = params[rowp + 80 + n];

        h = __expf(dt_t * A_dn) * h + (dt_t * u_t) * Bn;

        float yv = h * Cn;
        yv += __shfl_xor(yv, 1, 16);
        yv += __shfl_xor(yv, 2, 16);
        yv += __shfl_xor(yv, 4, 16);
        yv += __shfl_xor(yv, 8, 16);

        if (n == 0) {
            const float g = xz[((size_t)(bb * 2048 + t)) * 4096 + 2048 + d];
            yfull[rowud] = (yv + u_t * D_d) * silu_f(g);
        }
        if ((t & 63) == 0 && t + 64 < 2048) {   // global_prefetch_b8 hints
            __builtin_prefetch(&dtb[rowud + (size_t)64 * 2048], 0, 1);
            __builtin_prefetch(&u[rowud + (size_t)64 * 2048], 0, 1);
        }
    }
}

// ---------------------------------------------------------------------------
extern "C" void kernel_launch(void* const* d_in, const int* in_sizes, int n_in,
                              void* d_out, int out_size, void* d_ws, size_t ws_size,
                              hipStream_t stream)
{
    const float* x      = (const float*)d_in[0];
    const float* W_in   = (const float*)d_in[1];
    const float* conv_w = (const float*)d_in[2];
    const float* W_ssm  = (const float*)d_in[3];
    const float* W_dt   = (const float*)d_in[4];
    const float* b_dt   = (const float*)d_in[5];
    const float* log_A  = (const float*)d_in[6];
    const float* Dv     = (const float*)d_in[7];
    const float* W_out  = (const float*)d_in[8];
    float* out = (float*)d_out;

    const int M = 4096;  // b * L tokens

    // Workspace carving (floats): xz | u | params | dt   (~130 MB total)
    float* xz     = (float*)d_ws;                 // 4096 * 4096
    float* u      = xz + (size_t)M * 4096;        // 4096 * 2048
    float* params = u + (size_t)M * 2048;         // 4096 * 96
    float* dtb    = params + (size_t)M * 96;      // 4096 * 2048
    float* yfull  = u;                            // alias (safe: see scan)

    dim3 blk(256);

    // 1) xz = x @ W_in^T            (M=4096, N=4096, K=1024)
    wmma_gemm_nt<0><<<dim3(4096 / TILE_N, M / TILE_M), blk, 0, stream>>>(
        x, 1024, W_in, 1024, nullptr, xz, 4096, 1024);

    // 2) depthwise causal conv + SiLU
    conv_silu_kernel<<<dim3((M * 2048) / 256), blk, 0, stream>>>(xz, conv_w, u);

    // 3) params = u @ W_ssm^T       (N=96, K=2048)
    wmma_gemm_nt<0><<<dim3(96 / TILE_N, M / TILE_M), blk, 0, stream>>>(
        u, 2048, W_ssm, 2048, nullptr, params, 96, 2048);

    // 4) dt = softplus(d_low @ W_dt^T + b_dt)   (A = params[:, :64], K=64)
    wmma_gemm_nt<1><<<dim3(2048 / TILE_N, M / TILE_M), blk, 0, stream>>>(
        params, 96, W_dt, 64, b_dt, dtb, 2048, 64);

    // 5) selective scan + u*D + silu(gate) fusion
    scan_kernel<<<dim3(256), blk, 0, stream>>>(u, dtb, params, xz, log_A, Dv, yfull);

    // 6) out = yfull @ W_out^T      (N=1024, K=2048)
    wmma_gemm_nt<0><<<dim3(1024 / TILE_N, M / TILE_M), blk, 0, stream>>>(
        yfull, 2048, W_out, 2048, nullptr, out, 1024, 2048);
}